// QuantumLayer_72859825209391
// MI455X (gfx1250) — compile-verified
//
#include <hip/hip_runtime.h>

typedef float v2f __attribute__((ext_vector_type(2)));
typedef float v8f __attribute__((ext_vector_type(8)));

#define N_QUBITS 4
#define N_LAYERS 5

// ---------------------------------------------------------------------------
// Kernel 1: compose the full 16x16 complex circuit unitary from weights.
// One wave; thread t (t<16) evolves basis column |t>. Gates act column-wise,
// so no synchronization is needed. Result stored row-major: ur[i*16+j]=Re U[i][j].
// ---------------------------------------------------------------------------
__global__ void build_unitary_kernel(const float* __restrict__ w,
                                     float* __restrict__ ur,
                                     float* __restrict__ ui) {
    int t = threadIdx.x;
    __shared__ float sre[16][17];
    __shared__ float sim[16][17];
    if (t < 16) {
        for (int i = 0; i < 16; ++i) { sre[t][i] = (i == t) ? 1.0f : 0.0f; sim[t][i] = 0.0f; }
        for (int l = 0; l < N_LAYERS; ++l) {
            // Rot(phi,theta,omega) on each qubit (axis bit = 8>>q)
            for (int q = 0; q < N_QUBITS; ++q) {
                float phi = w[(l * 4 + q) * 3 + 0];
                float th  = w[(l * 4 + q) * 3 + 1];
                float om  = w[(l * 4 + q) * 3 + 2];
                float ct = cosf(0.5f * th), st = sinf(0.5f * th);
                float sa, ca, sb, cb;
                sincosf(0.5f * (phi + om), &sa, &ca);   // alpha
                sincosf(0.5f * (phi - om), &sb, &cb);   // beta
                // R00=ct e^{-ia}, R01=-st e^{ib}, R10=st e^{-ib}, R11=ct e^{ia}
                float r00r =  ct * ca, r00i = -ct * sa;
                float r01r = -st * cb, r01i = -st * sb;
                float r10r =  st * cb, r10i = -st * sb;
                float r11r =  ct * ca, r11i =  ct * sa;
                int mask = 8 >> q;
                for (int i = 0; i < 16; ++i) {
                    if (i & mask) continue;
                    int j = i | mask;
                    float ar = sre[t][i], ai = sim[t][i];
                    float br = sre[t][j], bi = sim[t][j];
                    sre[t][i] = r00r * ar - r00i * ai + r01r * br - r01i * bi;
                    sim[t][i] = r00r * ai + r00i * ar + r01r * bi + r01i * br;
                    sre[t][j] = r10r * ar - r10i * ai + r11r * br - r11i * bi;
                    sim[t][j] = r10r * ai + r10i * ar + r11r * bi + r11i * br;
                }
            }
            // CNOT ring: control q, target (q + r) % 4, r = (l % 3) + 1
            int r = (l % 3) + 1;
            for (int q = 0; q < N_QUBITS; ++q) {
                int mc = 8 >> q;
                int mt = 8 >> ((q + r) & 3);
                for (int i = 0; i < 16; ++i) {
                    if ((i & mc) && !(i & mt)) {
                        int j = i | mt;
                        float tr = sre[t][i]; sre[t][i] = sre[t][j]; sre[t][j] = tr;
                        float ti = sim[t][i]; sim[t][i] = sim[t][j]; sim[t][j] = ti;
                    }
                }
            }
        }
        for (int i = 0; i < 16; ++i) {
            ur[i * 16 + t] = sre[t][i];   // column t is U|t> -> U[:, t]
            ui[i * 16 + t] = sim[t][i];
        }
    }
}

// ---------------------------------------------------------------------------
// Kernel 2: batched state evolution via WMMA f32 16x16x4.
// Each wave handles 16-row batch tiles:
//   Dr += S_tile(16x16) * Ur^T,  Di += S_tile * Ui^T   (4 chained k-steps each)
//   P = Dr^2 + Di^2  -> staged through per-wave LDS (DS ops are wave-in-order,
//   no barrier) -> in-lane Walsh partial sums -> one shfl_xor(16) per <Z_w>
//   -> coalesced float4 stores.
// A layout: lane L -> row M=L&15, VGPR v -> K = 2*(L>>4) + v (per k-step).
// B layout: lane L -> col N=L&15, VGPR v -> K = 2*(L>>4) + v.
// D layout: lane L -> N=L&15, VGPR v -> M = v + 8*(L>>4).
// ---------------------------------------------------------------------------
__global__ void qforward_wmma_kernel(const float* __restrict__ x,
                                     const float* __restrict__ ur,
                                     const float* __restrict__ ui,
                                     float* __restrict__ out,
                                     int nTiles) {
    const int lane = threadIdx.x & 31;
    const int wave = threadIdx.x >> 5;
    const int wavesPerBlock = blockDim.x >> 5;
    const int gwave = blockIdx.x * wavesPerBlock + wave;
    const int nWaves = gridDim.x * wavesPerBlock;
    const int n   = lane & 15;   // amplitude / column index (also row idx on reload)
    const int khi = lane >> 4;   // K half fed by this lane / amplitude half on reload

    __shared__ float stage[8][256];           // per-wave 16x16 f32 staging tile
    float* st = &stage[wave][0];

    // B operands (U^T), invariant across tiles: element t of step kk is
    // U[n, 4*kk + 2*khi + t]; pairs are contiguous -> 8-byte vector loads.
    v2f br[4], bi[4];
#pragma unroll
    for (int kk = 0; kk < 4; ++kk) {
        int off = n * 16 + 4 * kk + 2 * khi;
        br[kk] = *(const v2f*)(ur + off);
        bi[kk] = *(const v2f*)(ui + off);
    }

    for (int tile = gwave; tile < nTiles; tile += nWaves) {
        const int base = tile << 4;
        const float4 xv = *(const float4*)(x + (size_t)(base + n) * 4);
        float s0, c0, s1, c1, s2, c2, s3, c3;
        __sincosf(0.5f * xv.x, &s0, &c0);     // fast HW sin/cos: |arg| small, ~1e-6 err
        __sincosf(0.5f * xv.y, &s1, &c1);
        __sincosf(0.5f * xv.z, &s2, &c2);
        __sincosf(0.5f * xv.w, &s3, &c3);
        const float f2 = khi ? s2 : c2;       // qubit-2 bit == khi for this lane's K half

        // s[j], j = 4*kk + 2*khi + t : bits (q0,q1,q2,q3) = (kk>>1, kk&1, khi, t)
        v2f A[4];
#pragma unroll
        for (int kk = 0; kk < 4; ++kk) {
            float p = ((kk & 2) ? s0 : c0) * ((kk & 1) ? s1 : c1) * f2;
            A[kk].x = p * c3;
            A[kk].y = p * s3;
        }

        v8f dr = {};
        v8f di = {};
#pragma unroll
        for (int kk = 0; kk < 4; ++kk) {
            dr = __builtin_amdgcn_wmma_f32_16x16x4_f32(false, A[kk], false, br[kk],
                                                       (short)0, dr, false, false);
            di = __builtin_amdgcn_wmma_f32_16x16x4_f32(false, A[kk], false, bi[kk],
                                                       (short)0, di, false, false);
        }

        // P[row][i] staged to LDS: lane holds amplitude i=n of rows v, v+8.
#pragma unroll
        for (int v = 0; v < 8; ++v) {
            float p = dr[v] * dr[v] + di[v] * di[v];
            st[(v + 8 * khi) * 16 + n] = p;   // lanes 0-15 / 16-31: two 64B runs
        }

        // Reload transposed: lane handles row n, amplitude half khi (j = 0..7,
        // amplitude i = 8*khi + j). DS is in-order within a wave -> no barrier.
        const float4 pa = *(const float4*)(st + n * 16 + 8 * khi);
        const float4 pb = *(const float4*)(st + n * 16 + 8 * khi + 4);
        const float a0 = pa.x, a1 = pa.y, a2 = pa.z, a3 = pa.w;
        const float a4 = pb.x, a5 = pb.y, a6 = pb.z, a7 = pb.w;

        // Shared-partial Walsh sums over j (bit2 -> qubit1, bit1 -> qubit2, bit0 -> qubit3)
        const float s04 = a0 + a4, s15 = a1 + a5, s26 = a2 + a6, s37 = a3 + a7;
        const float d04 = a0 - a4, d15 = a1 - a5, d26 = a2 - a6, d37 = a3 - a7;
        const float u1 = (d04 + d15) + (d26 + d37);        // sign by bit2 of j
        const float e0 = s04 + s15, e1 = s26 + s37;
        const float g0 = s04 - s15, g1 = s26 - s37;
        const float T  = e0 + e1;                          // plain half-sum
        const float u2 = e0 - e1;                          // sign by bit1 of j
        const float u3 = g0 + g1;                          // sign by bit0 of j

        // Qubit0 sign is the amplitude half itself; combine halves across lane^16.
        const float t0 = khi ? -T : T;
        const float z0 = t0 + __shfl_xor(t0, 16, 32);
        const float z1 = u1 + __shfl_xor(u1, 16, 32);
        const float z2 = u2 + __shfl_xor(u2, 16, 32);
        const float z3 = u3 + __shfl_xor(u3, 16, 32);

        if (lane < 16) {
            float4 zo = {z0, z1, z2, z3};
            *(float4*)(out + (size_t)(base + n) * 4) = zo;   // 16B coalesced per row
        }
    }
}

// ---------------------------------------------------------------------------
extern "C" void kernel_launch(void* const* d_in, const int* in_sizes, int n_in,
                              void* d_out, int out_size, void* d_ws, size_t ws_size,
                              hipStream_t stream) {
    const float* x = (const float*)d_in[0];   // (B, 4) float32
    const float* w = (const float*)d_in[1];   // (5, 4, 3) float32
    float* out = (float*)d_out;               // (B, 4) float32
    float* ur = (float*)d_ws;                 // 256 floats
    float* ui = ur + 256;                     // 256 floats

    int B = in_sizes[0] / N_QUBITS;
    int nTiles = B >> 4;                      // B = 262144 -> 16384 tiles

    build_unitary_kernel<<<1, 32, 0, stream>>>(w, ur, ui);

    int threads = 256;                        // 8 waves/block
    int blocks  = 1024;                       // 8192 waves -> 2 tiles each
    qforward_wmma_kernel<<<blocks, threads, 0, stream>>>(x, ur, ui, out, nTiles);
}